// RankingCriterionBidir_86543591015116
// MI455X (gfx1250) — compile-verified
//
#include <hip/hip_runtime.h>

// RankingCriterionBidir for MI455X (gfx1250, wave32).
// scores: [512,512] f32, labels: [512,512] int64 (values 0/1).
// Output: single f32 scalar.
//
// Strategy: one 512-thread workgroup per task (512 row tasks + 512 col tasks).
// Scores + pos flags staged in LDS as float2 (broadcast ds_load_b64 inner loop).
// Cross-lane reductions use V_WMMA_F32_16X16X4_F32 as a wave32 reduction
// engine (A = 32 partials zero-padded to 16x4, B = ones).

#define MARGIN 0.2f
#define MDIM 512

typedef __attribute__((ext_vector_type(2))) float v2f;
typedef __attribute__((ext_vector_type(8))) float v8f;

// Sum 32 f32 values (one per lane of a wave32) via one WMMA.
// A layout (16x4 f32): lanes 0-15 -> {K0,K1}, lanes 16-31 -> {K2,K3}.
// With a = {x, 0}: row m of A sums to x[m] + x[m+16].
// D[m,n] = x[m] + x[m+16] for all n. Summing the 8 accumulator VGPRs gives
// sum_{m=0..7}(...) on lanes 0-15 and sum_{m=8..15}(...) on lanes 16-31,
// so total = value@lane0 + value@lane16. EXEC must be all ones (it is:
// every wave that reaches this helper has all 32 lanes active).
__device__ __forceinline__ float wave_reduce_sum_wmma(float x) {
  v2f a; a[0] = x;    a[1] = 0.0f;
  v2f b; b[0] = 1.0f; b[1] = 1.0f;
  v8f c = {};
  c = __builtin_amdgcn_wmma_f32_16x16x4_f32(
      /*neg_a=*/false, a, /*neg_b=*/false, b,
      /*c_mod=*/(short)0, c, /*reuse_a=*/false, /*reuse_b=*/false);
  float r = c[0] + c[1] + c[2] + c[3] + c[4] + c[5] + c[6] + c[7];
  return __shfl(r, 0, 32) + __shfl(r, 16, 32);
}

// One block per task. task < 512: row `task`. task >= 512: column `task-512`.
__global__ __launch_bounds__(MDIM) void rank_side_kernel(
    const float* __restrict__ scores,
    const int*   __restrict__ lab_lo,   // int64 labels viewed as int pairs (low dword)
    float* __restrict__ means,          // [1024]
    float* __restrict__ valids) {       // [1024]
  __shared__ float2 sp[MDIM];           // {score, pos_flag}
  __shared__ float  wtot[16];
  __shared__ int    wcnt[16];

  const int t    = threadIdx.x;
  const int task = blockIdx.x;
  // Row task: contiguous; col task: stride-512 gather (L2-resident, tiny data).
  const int idx = (task < MDIM) ? (task * MDIM + t) : (t * MDIM + (task - MDIM));

  const float s   = scores[idx];
  const int   lab = lab_lo[2 * idx];          // low 32 bits of int64 (value 0 or 1)
  const float p   = (lab == 1) ? 1.0f : 0.0f; // pos flag; neg flag = 1 - p

  sp[t] = make_float2(s, p);
  __syncthreads();

  // Thread t plays the role of negative index j:
  //   acc = sum_i pos_i * relu(MARGIN + s_j - s_i)
  float acc = 0.0f;
#pragma unroll 8
  for (int i = 0; i < MDIM; ++i) {
    float2 v = sp[i];                         // broadcast ds_load_b64
    float  d = MARGIN + s - v.x;
    acc = fmaf(fmaxf(d, 0.0f), v.y, acc);
  }
  const float partial = (1.0f - p) * acc;     // keep only neg j

  // Per-wave reduction of partials via WMMA; pos count via ballot/popc.
  const float        wt  = wave_reduce_sum_wmma(partial);
  const unsigned long long bal = __ballot(p > 0.5f);
  const int          cnt = __popcll(bal);

  const int wave = t >> 5;
  const int lane = t & 31;
  if (lane == 0) { wtot[wave] = wt; wcnt[wave] = cnt; }
  __syncthreads();

  if (t == 0) {
    float tot = 0.0f; int npos = 0;
    for (int w = 0; w < 16; ++w) { tot += wtot[w]; npos += wcnt[w]; }
    const float fcnt  = (float)npos * (float)(MDIM - npos);
    const bool  valid = fcnt > 0.0f;
    means[task]  = valid ? (tot / fcnt) : 0.0f;
    valids[task] = valid ? 1.0f : 0.0f;
  }
}

// Single wave32 finalize: sum 1024 means and 1024 valid flags, divide.
__global__ __launch_bounds__(32) void rank_final_kernel(
    const float* __restrict__ means,
    const float* __restrict__ valids,
    float* __restrict__ out) {
  const int lane = threadIdx.x;
  float sm = 0.0f, sv = 0.0f;
  for (int k = lane; k < 2 * MDIM; k += 32) {
    sm += means[k];
    sv += valids[k];
  }
  const float tm = wave_reduce_sum_wmma(sm);
  const float tv = wave_reduce_sum_wmma(sv);
  if (lane == 0) out[0] = tm / tv;
}

extern "C" void kernel_launch(void* const* d_in, const int* in_sizes, int n_in,
                              void* d_out, int out_size, void* d_ws, size_t ws_size,
                              hipStream_t stream) {
  (void)in_sizes; (void)n_in; (void)out_size; (void)ws_size;
  const float* scores = (const float*)d_in[0];
  const int*   lab_lo = (const int*)d_in[1];   // int64 -> read low dwords

  float* means  = (float*)d_ws;          // [1024]
  float* valids = means + 2 * MDIM;      // [1024]  (8 KB total workspace)

  rank_side_kernel<<<dim3(2 * MDIM), dim3(MDIM), 0, stream>>>(
      scores, lab_lo, means, valids);
  rank_final_kernel<<<dim3(1), dim3(32), 0, stream>>>(
      means, valids, (float*)d_out);
}